// StreamingHistogram_6811818131596
// MI455X (gfx1250) — compile-verified
//
#include <hip/hip_runtime.h>
#include <stdint.h>

// Problem geometry (from reference): x is [32, 12, 256, 256] fp32,
// hist_counts is [12, 2200] fp32. torch.histc semantics over [-1200, 1000].
#define BATCH      32
#define C_CHANNELS 12
#define BINS       2200
#define PLANE      65536            // 256*256 elements per (b, c) plane
#define SPLIT      2                // blocks per plane
#define CHUNK      (PLANE / SPLIT)  // 32768 elements per block
#define CHUNK_VEC  (CHUNK / 4)      // 8192 float4 per block
#define NWAVES     8                // 256 threads / wave32
#define WAVESZ     32
#define ITERS      (CHUNK_VEC / (NWAVES * WAVESZ)) // 32 pipeline steps/wave
#define DEPTH      8                // async pipeline depth (ASYNCcnt <= 7)

#define MIN_VALF  (-1200.0f)
#define MAX_VALF  (1000.0f)

// s_wait_asynccnt with a literal immediate (per-wave counter).
#define WAIT_ASYNC(n) asm volatile("s_wait_asynccnt " #n ::: "memory")

// Kernel 1: out = hist_counts (fresh every launch -> deterministic replays).
__global__ void StreamingHistogram_init_kernel(const float* __restrict__ hin,
                                               float* __restrict__ out, int n) {
    int i = blockIdx.x * blockDim.x + threadIdx.x;
    if (i < n) out[i] = hin[i];
}

// Kernel 2: per-block LDS-privatized histogram over one (b, c, half-plane)
// chunk, fed by an 8-deep global_load_async_to_lds_b128 pipeline per wave.
__global__ __launch_bounds__(256) void StreamingHistogram_main_kernel(
    const float* __restrict__ x, float* __restrict__ out) {
    __shared__ unsigned int hist[BINS];                 // 8.8 KB
    __shared__ float4 stage[NWAVES][DEPTH][WAVESZ];     // 32 KB staging

    const int tid = threadIdx.x;
    for (int b = tid; b < BINS; b += 256) hist[b] = 0u;
    __syncthreads();

    const int blk  = blockIdx.x;
    const int c    = (blk / SPLIT) % C_CHANNELS;
    const int bidx = blk / (SPLIT * C_CHANNELS);
    const int half = blk % SPLIT;

    // Uniform 64-bit base of this block's 128 KB chunk -> SGPR pair (SADDR).
    const unsigned long long base =
        (unsigned long long)(uintptr_t)x +
        ((unsigned long long)(bidx * C_CHANNELS + c) * PLANE +
         (unsigned long long)half * CHUNK) * sizeof(float);

    const int w = tid >> 5;   // wave id within block (wave32)
    const int l = tid & 31;   // lane id

    // LDS byte offsets (low 32 bits of the flat LDS address) for this lane's
    // DEPTH staging slots — these are the VDST operands of the async loads.
    uint32_t lds_off[DEPTH];
    #pragma unroll
    for (int d = 0; d < DEPTH; ++d)
        lds_off[d] = (uint32_t)(uintptr_t)&stage[w][d][l];

    auto issue = [&](int k) {
        // GVS addressing: mem = SADDR(64b) + VADDR(32b offset). Coalesced:
        // 32 lanes x 16B = one contiguous 512B wave transaction. Input is
        // read exactly once -> non-temporal hint keeps L2 clean.
        uint32_t goff = (uint32_t)(((k * NWAVES + w) * WAVESZ + l) * 16);
        asm volatile("global_load_async_to_lds_b128 %0, %1, %2 th:TH_LOAD_NT"
                     :
                     : "v"(lds_off[k & (DEPTH - 1)]), "v"(goff), "s"(base)
                     : "memory");
    };

    auto consume = [&](int k) {
        float4 v = stage[w][k & (DEPTH - 1)][l];
        float vals[4] = {v.x, v.y, v.z, v.w};
        #pragma unroll
        for (int j = 0; j < 4; ++j) {
            float t = vals[j];
            // torch.histc: ignore out-of-range (NaN also fails the compare);
            // scale = BINS / (MAX - MIN) == 1.0 here, kept symbolic.
            if (t >= MIN_VALF && t <= MAX_VALF) {
                int bin = (int)floorf((t - MIN_VALF) *
                                      ((float)BINS / (MAX_VALF - MIN_VALF)));
                bin = bin < 0 ? 0 : (bin > BINS - 1 ? BINS - 1 : bin);
                atomicAdd(&hist[bin], 1u);   // ds_add_u32 (no return)
            }
        }
    };

    // Prime the async pipeline (ITERS == 32 >= DEPTH == 8).
    #pragma unroll
    for (int k = 0; k < DEPTH; ++k) issue(k);

    // Steady state: ASYNCcnt is per-wave and async loads complete in order,
    // so s_wait_asynccnt DEPTH-1 guarantees the oldest tile has landed in
    // LDS. consume(k)'s ds_load data is consumed (compiler-inserted dscnt
    // wait) before issue(k+DEPTH) re-targets the same buffer slot, so the
    // 8-slot ring is race-free within the wave.
    for (int k = 0; k < ITERS - (DEPTH - 1); ++k) {
        WAIT_ASYNC(7);
        consume(k);
        if (k + DEPTH < ITERS) issue(k + DEPTH);
    }
    // Drain epilogue.
    WAIT_ASYNC(6); consume(ITERS - 7);
    WAIT_ASYNC(5); consume(ITERS - 6);
    WAIT_ASYNC(4); consume(ITERS - 5);
    WAIT_ASYNC(3); consume(ITERS - 4);
    WAIT_ASYNC(2); consume(ITERS - 3);
    WAIT_ASYNC(1); consume(ITERS - 2);
    WAIT_ASYNC(0); consume(ITERS - 1);

    __syncthreads();

    // Flush the private histogram. Counts are integers < 2^24, so fp32
    // atomic accumulation is exact and order-independent (deterministic).
    float* outc = out + c * BINS;
    for (int b = tid; b < BINS; b += 256) {
        unsigned int cnt = hist[b];
        if (cnt) atomicAdd(&outc[b], (float)cnt);   // global_atomic_add_f32
    }
}

extern "C" void kernel_launch(void* const* d_in, const int* in_sizes, int n_in,
                              void* d_out, int out_size, void* d_ws, size_t ws_size,
                              hipStream_t stream) {
    const float* x       = (const float*)d_in[0];   // [32,12,256,256] fp32
    const float* hist_in = (const float*)d_in[1];   // [12,2200] fp32
    float* out = (float*)d_out;                     // [12,2200] fp32

    (void)in_sizes; (void)n_in; (void)d_ws; (void)ws_size;

    StreamingHistogram_init_kernel<<<(out_size + 255) / 256, 256, 0, stream>>>(
        hist_in, out, out_size);

    StreamingHistogram_main_kernel<<<BATCH * C_CHANNELS * SPLIT, 256, 0, stream>>>(
        x, out);
}